// GATLayer_19121194402173
// MI455X (gfx1250) — compile-verified
//
#include <hip/hip_runtime.h>
#include <hip/hip_bf16.h>

// ---------------------------------------------------------------------------
// GAT layer for MI455X (gfx1250, wave32).
// Projection GEMM runs on V_WMMA_F32_16X16X4_F32 (full f32 precision; the
// kernel is L2/atomic bound so there is no reason to quantize).
// ---------------------------------------------------------------------------

#define N_NODES 50000
#define N_EDGES 800000
#define C_IN    128
#define N_HEAD  4
#define C_OUT   32
#define C_TOT   (N_HEAD * C_OUT)     // 128 = flattened output columns
#define NEG_SLOPE 0.2f

typedef __attribute__((ext_vector_type(2))) float v2f;
typedef __attribute__((ext_vector_type(8))) float v8f;

// ---------------------------------------------------------------------------
// Kernel 1: h_proj[n, h*32+o] = sum_i h[n,i] * W[h,i,o]
// Viewed as a single (N x 128) @ (128 x 128) GEMM where
//   B[k, c] = W[c/32, k, c%32].
// Block = 256 threads = 8 waves. Each block: 16-row M tile.
// Wave w computes the 16x16 output tile at columns [16w, 16w+16).
// B (all of W, 64 KB) is staged in LDS once per block.
// ---------------------------------------------------------------------------
__global__ __launch_bounds__(256)
void gat_proj_wmma_f32(const float* __restrict__ h,
                       const float* __restrict__ W,
                       float* __restrict__ h_proj) {
    __shared__ float Bs[C_IN * C_TOT];          // 64 KB (WGP has 320 KB)

    const int tid  = threadIdx.x;
    const int wave = tid >> 5;                  // 0..7
    const int lane = tid & 31;                  // wave32
    const int rowBase = blockIdx.x * 16;        // 50000 / 16 = 3125 blocks exactly

    // Stage B into LDS: Bs[k*128 + c] = W[(c>>5)*IN*OUT + k*OUT + (c&31)]
    for (int i = tid; i < C_IN * C_TOT; i += 256) {
        const int k = i >> 7;
        const int c = i & 127;
        Bs[i] = W[(c >> 5) * (C_IN * C_OUT) + k * C_OUT + (c & 31)];
    }
    __syncthreads();

    // f32 WMMA 16x16x4 fragment layout (ISA 7.12.2):
    //  A: lane l -> M = l&15 ; VGPR0/1 = A[M][k0 + 2*(l>>4) + {0,1}]
    //  B: lane l -> N = l&15 ; VGPR0/1 = B[k0 + 2*(l>>4) + {0,1}][N]
    const int mrow  = lane & 15;
    const int khalf = (lane >> 4) << 1;         // 0 or 2
    const int col   = (wave << 4) + (lane & 15);

    const float* arow = h + (size_t)(rowBase + mrow) * C_IN;

    v8f acc = {};
    #pragma unroll
    for (int k0 = 0; k0 < C_IN; k0 += 4) {
        // A fragment: contiguous pair -> 8-byte global load
        v2f a = *(const v2f*)(arow + k0 + khalf);
        // B fragment from LDS
        v2f b;
        b.x = Bs[(k0 + khalf)     * C_TOT + col];
        b.y = Bs[(k0 + khalf + 1) * C_TOT + col];
        acc = __builtin_amdgcn_wmma_f32_16x16x4_f32(
            /*neg_a=*/false, a, /*neg_b=*/false, b,
            /*c_mod=*/(short)0, acc, /*reuse_a=*/false, /*reuse_b=*/false);
    }

    // D layout: VGPR r, lanes 0-15 -> M=r, lanes 16-31 -> M=8+r; N = lane&15
    const int orow = rowBase + ((lane >> 4) << 3);
    #pragma unroll
    for (int r = 0; r < 8; ++r) {
        h_proj[(size_t)(orow + r) * C_TOT + col] = acc[r];
    }
}

// ---------------------------------------------------------------------------
// Kernel 0: initialize accumulators every launch (graph-replay deterministic).
// ---------------------------------------------------------------------------
__global__ void gat_init(float* __restrict__ out,     // N*128, zeroed
                         float* __restrict__ smax,    // N*4  -> -1e9
                         float* __restrict__ denom) { // N*4  -> 0
    const int idx = blockIdx.x * blockDim.x + threadIdx.x;
    if (idx < N_NODES * C_TOT) out[idx] = 0.0f;
    if (idx < N_NODES * N_HEAD) {
        smax[idx]  = -1000000000.0f;   // matches reference's -inf -> -1e9 mapping
        denom[idx] = 0.0f;
    }
}

// ---------------------------------------------------------------------------
// Kernel 2: per-(node,head) attention logits s_src / s_dst.
// ---------------------------------------------------------------------------
__global__ void gat_node_scores(const float* __restrict__ h_proj,
                                const float* __restrict__ a,
                                float* __restrict__ s_src,
                                float* __restrict__ s_dst) {
    const int idx = blockIdx.x * blockDim.x + threadIdx.x;
    if (idx >= N_NODES * N_HEAD) return;
    const int n  = idx >> 2;
    const int hh = idx & 3;
    const float* hp = h_proj + (size_t)n * C_TOT + hh * C_OUT;
    const float* as = a + hh * (2 * C_OUT);
    const float* ad = as + C_OUT;
    float accs = 0.0f, accd = 0.0f;
    #pragma unroll
    for (int o = 0; o < C_OUT; ++o) {
        const float v = hp[o];
        accs += v * as[o];
        accd += v * ad[o];
    }
    s_src[idx] = accs;
    s_dst[idx] = accd;
}

// ---------------------------------------------------------------------------
// Kernel 3: edge scores (leaky relu) + segment max via f32 atomic max.
// ---------------------------------------------------------------------------
__global__ void gat_edge_score(const int* __restrict__ src,
                               const int* __restrict__ dst,
                               const float* __restrict__ s_src,
                               const float* __restrict__ s_dst,
                               float* __restrict__ score,
                               float* __restrict__ smax) {
    const int e = blockIdx.x * blockDim.x + threadIdx.x;
    if (e >= N_EDGES) return;
    const int s = src[e];
    const int d = dst[e];
    #pragma unroll
    for (int hh = 0; hh < N_HEAD; ++hh) {
        float sc = s_src[s * N_HEAD + hh] + s_dst[d * N_HEAD + hh];
        sc = sc > 0.0f ? sc : NEG_SLOPE * sc;
        score[e * N_HEAD + hh] = sc;
        atomicMax(&smax[d * N_HEAD + hh], sc);   // global_atomic_max_num_f32
    }
}

// ---------------------------------------------------------------------------
// Kernel 4: score_exp = exp(score - smax[dst]); denom += score_exp (atomic).
// ---------------------------------------------------------------------------
__global__ void gat_edge_exp(const int* __restrict__ dst,
                             const float* __restrict__ smax,
                             float* __restrict__ score,
                             float* __restrict__ denom) {
    const int idx = blockIdx.x * blockDim.x + threadIdx.x;
    if (idx >= N_EDGES * N_HEAD) return;
    const int e  = idx >> 2;
    const int hh = idx & 3;
    const int d  = dst[e];
    const float ex = __expf(score[idx] - smax[d * N_HEAD + hh]);
    score[idx] = ex;
    atomicAdd(&denom[d * N_HEAD + hh], ex);
}

// ---------------------------------------------------------------------------
// Kernel 5: alpha = score_exp / max(denom[dst], 1e-16)   (in place)
// ---------------------------------------------------------------------------
__global__ void gat_alpha(const int* __restrict__ dst,
                          const float* __restrict__ denom,
                          float* __restrict__ score) {
    const int idx = blockIdx.x * blockDim.x + threadIdx.x;
    if (idx >= N_EDGES * N_HEAD) return;
    const int e  = idx >> 2;
    const int hh = idx & 3;
    const int d  = dst[e];
    score[idx] = score[idx] / fmaxf(denom[d * N_HEAD + hh], 1e-16f);
}

// ---------------------------------------------------------------------------
// Kernel 6: out[dst, c] += alpha[e, c/32] * h_proj[src, c]   (atomic f32 add)
// 128 threads per edge: each wave does a contiguous 128 B gather of
// h_proj[src] and contiguous atomics into out[dst] -- all L2-resident.
// ---------------------------------------------------------------------------
__global__ void gat_aggregate(const int* __restrict__ src,
                              const int* __restrict__ dst,
                              const float* __restrict__ alpha,
                              const float* __restrict__ h_proj,
                              float* __restrict__ out) {
    const int idx = blockIdx.x * blockDim.x + threadIdx.x;   // E*128 = 102.4M
    if (idx >= N_EDGES * C_TOT) return;
    const int e  = idx >> 7;
    const int c  = idx & 127;
    const int hh = c >> 5;
    const int s  = src[e];
    const int d  = dst[e];
    const float v = alpha[e * N_HEAD + hh] * h_proj[(size_t)s * C_TOT + c];
    atomicAdd(&out[(size_t)d * C_TOT + c], v);
}

// ---------------------------------------------------------------------------
// Kernel 7: ELU in place.
// ---------------------------------------------------------------------------
__global__ void gat_elu(float* __restrict__ out) {
    const int idx = blockIdx.x * blockDim.x + threadIdx.x;
    if (idx >= N_NODES * C_TOT) return;
    const float x = out[idx];
    out[idx] = x > 0.0f ? x : (__expf(x) - 1.0f);
}

// ---------------------------------------------------------------------------
// Host-side launcher.
// ---------------------------------------------------------------------------
extern "C" void kernel_launch(void* const* d_in, const int* in_sizes, int n_in,
                              void* d_out, int out_size, void* d_ws, size_t ws_size,
                              hipStream_t stream) {
    const float* h   = (const float*)d_in[0];            // N x 128
    const int*   ei  = (const int*)d_in[1];              // 2 x E
    const float* W   = (const float*)d_in[2];            // H x 128 x 32
    const float* a   = (const float*)d_in[3];            // H x 64
    float*       out = (float*)d_out;                    // N x 128

    const int* src = ei;
    const int* dst = ei + N_EDGES;

    // Workspace layout (floats)
    float* ws      = (float*)d_ws;
    float* h_proj  = ws;                                  // N*128 = 6,400,000
    float* s_src   = h_proj + (size_t)N_NODES * C_TOT;    // N*4
    float* s_dst   = s_src  + (size_t)N_NODES * N_HEAD;   // N*4
    float* smax    = s_dst  + (size_t)N_NODES * N_HEAD;   // N*4
    float* denom   = smax   + (size_t)N_NODES * N_HEAD;   // N*4
    float* score   = denom  + (size_t)N_NODES * N_HEAD;   // E*4 = 3,200,000

    const int B = 256;

    gat_init<<<(N_NODES * C_TOT + B - 1) / B, B, 0, stream>>>(out, smax, denom);

    gat_proj_wmma_f32<<<N_NODES / 16, 256, 0, stream>>>(h, W, h_proj);

    gat_node_scores<<<(N_NODES * N_HEAD + B - 1) / B, B, 0, stream>>>(
        h_proj, a, s_src, s_dst);

    gat_edge_score<<<(N_EDGES + B - 1) / B, B, 0, stream>>>(
        src, dst, s_src, s_dst, score, smax);

    gat_edge_exp<<<(N_EDGES * N_HEAD + B - 1) / B, B, 0, stream>>>(
        dst, smax, score, denom);

    gat_alpha<<<(N_EDGES * N_HEAD + B - 1) / B, B, 0, stream>>>(
        dst, denom, score);

    gat_aggregate<<<(N_EDGES * C_TOT + B - 1) / B, B, 0, stream>>>(
        src, dst, score, h_proj, out);

    gat_elu<<<(N_NODES * C_TOT + B - 1) / B, B, 0, stream>>>(out);
}